// DESIGN_52398601011578
// MI455X (gfx1250) — compile-verified
//
#include <hip/hip_runtime.h>
#include <hip/hip_bf16.h>

// ---------------------------------------------------------------------------
// Problem constants (from the reference)
// ---------------------------------------------------------------------------
#define N_USERS 10000
#define N_ITEMS 40000
#define N_TOT   (N_USERS + N_ITEMS)
#define HID     64
#define NNZ     2000000
#define BATCH   8192
#define HOP     3

typedef __attribute__((ext_vector_type(2))) float v2f;
typedef __attribute__((ext_vector_type(8))) float v8f;

// ---------------------------------------------------------------------------
// Dense social GCN hop:   Hn = S @ Hc ;  Acc += Hn
// S: [N_USERS, N_USERS] fp32, Hc/Hn/Acc: [N_USERS, HID]
// grid.x = N_USERS/16 = 625 M-tiles, block = 128 threads (4 wave32).
// Each wave covers K-range of 2500 and the full N=64 strip with 4 WMMA tiles.
// Partials reduced across the 4 waves through LDS.
// ---------------------------------------------------------------------------
__global__ __launch_bounds__(128) void sgr_social_hop(
    const float* __restrict__ S, const float* __restrict__ Hc,
    float* __restrict__ Hn, float* __restrict__ Acc) {
  __shared__ float red[4][32][32];  // [wave][lane][4 tiles * 8 vgprs] = 16KB

  const int tid    = threadIdx.x;
  const int wave   = tid >> 5;
  const int lane   = tid & 31;
  const int laneHi = lane >> 4;   // 0/1 : selects K pair (A) / K row pair (B)
  const int laneLo = lane & 15;   // M for A-frag, N-within-tile for B/C frags
  const int m0     = blockIdx.x * 16;

  v8f c0 = {}, c1 = {}, c2 = {}, c3 = {};

  const int kBegin = wave * (N_USERS / 4);
  const int kEnd   = kBegin + (N_USERS / 4);

  // A-matrix 16x4 fp32 layout: VGPR0 = K = 2*laneHi, VGPR1 = K+1, row M = laneLo
  const float* Srow = S + (size_t)(m0 + laneLo) * N_USERS;

  for (int k0 = kBegin; k0 < kEnd; k0 += 4) {
    const float* ap = Srow + k0 + 2 * laneHi;
    // Streaming read of S (400MB > L2): non-temporal, plus prefetch ahead.
    v2f a = __builtin_nontemporal_load((const v2f*)ap);
    __builtin_prefetch(ap + 128, 0, 1);

    // B-matrix 4x16 fp32 layout: VGPR0 row K = k0+2*laneHi, VGPR1 row K+1,
    // column = tile*16 + laneLo.  Four N-tiles cover HID=64.
    const float* bp = Hc + (size_t)(k0 + 2 * laneHi) * HID + laneLo;
    v2f b0 = {bp[0],  bp[64 + 0]};
    v2f b1 = {bp[16], bp[64 + 16]};
    v2f b2 = {bp[32], bp[64 + 32]};
    v2f b3 = {bp[48], bp[64 + 48]};

    c0 = __builtin_amdgcn_wmma_f32_16x16x4_f32(false, a, false, b0, (short)0, c0, false, false);
    c1 = __builtin_amdgcn_wmma_f32_16x16x4_f32(false, a, false, b1, (short)0, c1, false, false);
    c2 = __builtin_amdgcn_wmma_f32_16x16x4_f32(false, a, false, b2, (short)0, c2, false, false);
    c3 = __builtin_amdgcn_wmma_f32_16x16x4_f32(false, a, false, b3, (short)0, c3, false, false);
  }

  // Spill per-wave partials to LDS.
  float* myred = &red[wave][lane][0];
#pragma unroll
  for (int i = 0; i < 8; ++i) {
    myred[i]      = c0[i];
    myred[8 + i]  = c1[i];
    myred[16 + i] = c2[i];
    myred[24 + i] = c3[i];
  }
  __syncthreads();

  // Cooperative cross-wave reduction of the 16x64 tile (8 outputs/thread).
  // C/D layout: VGPR r holds M=r (lanes 0-15) and M=r+8 (lanes 16-31), N=lane&15.
  for (int o = tid; o < 16 * HID; o += 128) {
    int m = o >> 6;
    int n = o & 63;
    int t = n >> 4;
    int l = ((m & 8) ? 16 : 0) + (n & 15);
    int r = (t << 3) + (m & 7);
    float s = red[0][l][r] + red[1][l][r] + red[2][l][r] + red[3][l][r];
    size_t idx = (size_t)(m0 + m) * HID + n;
    Hn[idx] = s;
    Acc[idx] += s;
  }
}

// ---------------------------------------------------------------------------
// Utility kernels
// ---------------------------------------------------------------------------
__global__ void sgr_copy2(const float4* __restrict__ src, float4* __restrict__ a,
                          float4* __restrict__ b, int n4) {
  int i = blockIdx.x * blockDim.x + threadIdx.x;
  if (i < n4) {
    float4 v = src[i];
    a[i] = v;
    b[i] = v;
  }
}

__global__ void sgr_zero_i(int* __restrict__ p, int n) {
  int i = blockIdx.x * blockDim.x + threadIdx.x;
  if (i < n) p[i] = 0;
}

// ---------------------------------------------------------------------------
// CSR build (once per launch; A_rows/cols/vals are constant across all hops)
// ---------------------------------------------------------------------------
__global__ void sgr_hist(const int* __restrict__ rows, int* __restrict__ cnt, int nnz) {
  int i = blockIdx.x * blockDim.x + threadIdx.x;
  if (i < nnz) atomicAdd(&cnt[rows[i]], 1);
}

#define SCAN_T 1024
__global__ __launch_bounds__(SCAN_T) void sgr_scan(const int* __restrict__ cnt,
                                                   int* __restrict__ rptr, int n) {
  __shared__ int part[SCAN_T];
  const int t = threadIdx.x;
  const int chunk = (n + SCAN_T - 1) / SCAN_T;
  const int b = t * chunk;
  const int e = min(b + chunk, n);
  int s = 0;
  for (int i = b; i < e; ++i) s += cnt[i];
  part[t] = s;
  __syncthreads();
  // Hillis-Steele inclusive scan over the 1024 partials
  for (int off = 1; off < SCAN_T; off <<= 1) {
    int v = (t >= off) ? part[t - off] : 0;
    __syncthreads();
    part[t] += v;
    __syncthreads();
  }
  int run = (t == 0) ? 0 : part[t - 1];  // exclusive prefix for this chunk
  for (int i = b; i < e; ++i) {
    rptr[i] = run;
    run += cnt[i];
  }
  if (t == SCAN_T - 1) rptr[n] = run;
}

__global__ void sgr_csr_fill(const int* __restrict__ rows, const int* __restrict__ cols,
                             const float* __restrict__ vals, const int* __restrict__ rptr,
                             int* __restrict__ cursor, int* __restrict__ ecol,
                             float* __restrict__ eval, int nnz) {
  int i = blockIdx.x * blockDim.x + threadIdx.x;
  if (i >= nnz) return;
  int r = rows[i];
  int p = rptr[r] + atomicAdd(&cursor[r], 1);
  ecol[p] = cols[i];
  eval[p] = vals[i];
}

// ---------------------------------------------------------------------------
// CSR SpMV hop:  Hn[r,:] = sum_e val[e]*Hc[col[e],:] ;  Acc[r,:] += Hn[r,:]
// One wave per row; lane owns 2 features; edge metadata broadcast via __shfl
// (ds_bpermute lane path) so each edge's col/val is loaded once per wave.
// No float atomics, no zero pass, Acc update fused.
// ---------------------------------------------------------------------------
__global__ __launch_bounds__(256) void sgr_spmm_csr(
    const int* __restrict__ rptr, const int* __restrict__ ecol,
    const float* __restrict__ eval, const float* __restrict__ Hc,
    float* __restrict__ Hn, float* __restrict__ Acc) {
  const int row  = (blockIdx.x * blockDim.x + threadIdx.x) >> 5;
  const int lane = threadIdx.x & 31;
  if (row >= N_TOT) return;
  const int e0 = rptr[row];
  const int e1 = rptr[row + 1];
  const int f  = lane * 2;

  float ax = 0.f, ay = 0.f;
  for (int base = e0; base < e1; base += 32) {
    int ee = base + lane;
    int cv = (ee < e1) ? ecol[ee] : 0;
    float vv = (ee < e1) ? eval[ee] : 0.f;
    int cnt = min(32, e1 - base);
    for (int j = 0; j < cnt; ++j) {
      int c = __shfl(cv, j);
      float v = __shfl(vv, j);
      const float2 h = *(const float2*)(Hc + (size_t)c * HID + f);
      ax += v * h.x;
      ay += v * h.y;
    }
  }
  const size_t o = (size_t)row * HID + f;
  *(float2*)(Hn + o) = make_float2(ax, ay);
  float2 a = *(const float2*)(Acc + o);
  a.x += ax;
  a.y += ay;
  *(float2*)(Acc + o) = a;
}

// ---------------------------------------------------------------------------
// Output gathers
// ---------------------------------------------------------------------------
__global__ void sgr_gather(float* __restrict__ dst, const float* __restrict__ src,
                           const int* __restrict__ idx, int rowOff, float scale) {
  int t = blockIdx.x * blockDim.x + threadIdx.x;  // BATCH*64 threads
  int i = t >> 6;
  int f = t & 63;
  int r = idx[i] + rowOff;
  dst[t] = scale * src[(size_t)r * HID + f];
}

__global__ void sgr_gather2(float* __restrict__ dst, const float* __restrict__ sa,
                            const float* __restrict__ sb, const int* __restrict__ idx,
                            float scale) {
  int t = blockIdx.x * blockDim.x + threadIdx.x;
  int i = t >> 6;
  int f = t & 63;
  size_t o = (size_t)idx[i] * HID + f;
  dst[t] = scale * (sa[o] + sb[o]);
}

__global__ void sgr_scale_copy(float* __restrict__ dst, const float* __restrict__ src,
                               float scale, int n) {
  int t = blockIdx.x * blockDim.x + threadIdx.x;
  if (t < n) dst[t] = scale * src[t];
}

// ---------------------------------------------------------------------------
// Host-side orchestration
// ---------------------------------------------------------------------------
extern "C" void kernel_launch(void* const* d_in, const int* in_sizes, int n_in,
                              void* d_out, int out_size, void* d_ws, size_t ws_size,
                              hipStream_t stream) {
  const int*   users = (const int*)d_in[0];
  const int*   pos   = (const int*)d_in[1];
  const int*   neg   = (const int*)d_in[2];
  const float* uemb  = (const float*)d_in[3];
  const float* iemb  = (const float*)d_in[4];
  const float* uemb1 = (const float*)d_in[5];
  const float* iemb1 = (const float*)d_in[6];
  const float* uemb2 = (const float*)d_in[7];
  const float* iemb2 = (const float*)d_in[8];
  const float* S     = (const float*)d_in[9];
  const int*   Ar    = (const int*)d_in[10];
  const int*   Ac    = (const int*)d_in[11];
  const float* Av    = (const float*)d_in[12];
  (void)in_sizes; (void)n_in; (void)ws_size; (void)out_size;

  const int SOC_N = N_USERS * HID;   // 640,000 floats
  const int UI_N  = N_TOT * HID;     // 3,200,000 floats

  float* ws      = (float*)d_ws;
  float* socCur  = ws;
  float* socNxt  = socCur + SOC_N;
  float* socAccA = socNxt + SOC_N;
  float* socAccB = socAccA + SOC_N;
  float* uiCur   = socAccB + SOC_N;
  float* uiNxt   = uiCur + UI_N;
  float* uiAccA  = uiNxt + UI_N;
  float* uiAccR  = uiAccA + UI_N;
  float* evalBuf = uiAccR + UI_N;            // NNZ floats
  int*   ecolBuf = (int*)(evalBuf + NNZ);    // NNZ ints
  int*   rptr    = ecolBuf + NNZ;            // N_TOT+1 ints
  int*   cntBuf  = rptr + (N_TOT + 1);       // N_TOT ints
  int*   curBuf  = cntBuf + N_TOT;           // N_TOT ints

  const float invL = 1.0f / (HOP + 1);   // 0.25

  // ---- CSR build (A graph is constant across both ui_gcn runs) ----
  sgr_zero_i<<<(2 * N_TOT + 255) / 256, 256, 0, stream>>>(cntBuf, 2 * N_TOT);
  sgr_hist<<<(NNZ + 255) / 256, 256, 0, stream>>>(Ar, cntBuf, NNZ);
  sgr_scan<<<1, SCAN_T, 0, stream>>>(cntBuf, rptr, N_TOT);
  sgr_csr_fill<<<(NNZ + 255) / 256, 256, 0, stream>>>(Ar, Ac, Av, rptr, curBuf,
                                                      ecolBuf, evalBuf, NNZ);

  // ---- social_gcn runs (dense WMMA GEMM chain) ----
  auto run_social = [&](const float* emb, float* acc) {
    sgr_copy2<<<(SOC_N / 4 + 255) / 256, 256, 0, stream>>>(
        (const float4*)emb, (float4*)socCur, (float4*)acc, SOC_N / 4);
    float* cur = socCur;
    float* nxt = socNxt;
    for (int h = 0; h < HOP; ++h) {
      sgr_social_hop<<<N_USERS / 16, 128, 0, stream>>>(S, cur, nxt, acc);
      float* t = cur; cur = nxt; nxt = t;
    }
  };
  run_social(uemb,  socAccA);   // all_user_S   (pre-scale)
  run_social(uemb1, socAccB);   // all_user_social (pre-scale)

  // ---- ui_gcn runs (CSR SpMV chain, no float atomics) ----
  auto run_ui = [&](const float* ue, const float* ie, float* acc) {
    sgr_copy2<<<(SOC_N / 4 + 255) / 256, 256, 0, stream>>>(
        (const float4*)ue, (float4*)uiCur, (float4*)acc, SOC_N / 4);
    const int ITEM_N = N_ITEMS * HID;
    sgr_copy2<<<(ITEM_N / 4 + 255) / 256, 256, 0, stream>>>(
        (const float4*)ie, (float4*)(uiCur + SOC_N), (float4*)(acc + SOC_N), ITEM_N / 4);
    float* cur = uiCur;
    float* nxt = uiNxt;
    for (int h = 0; h < HOP; ++h) {
      sgr_spmm_csr<<<(N_TOT * 32 + 255) / 256, 256, 0, stream>>>(rptr, ecolBuf, evalBuf,
                                                                 cur, nxt, acc);
      float* t = cur; cur = nxt; nxt = t;
    }
  };
  run_ui(uemb,  iemb,  uiAccA);   // all_user_A / all_item   (pre-scale)
  run_ui(uemb2, iemb2, uiAccR);   // all_user_rating / all_item_rating (pre-scale)

  // ---- outputs (return-order concatenation) ----
  float* out = (float*)d_out;
  const int GB = BATCH * HID;  // 524,288 per gathered section
  float* o0  = out;                 // all_user[users]
  float* o1  = o0 + GB;             // all_item[pos]
  float* o2  = o1 + GB;             // all_item[neg]
  float* o3  = o2 + GB;             // all_user_S          [N_USERS,HID]
  float* o4  = o3 + SOC_N;          // all_user_A          [N_USERS,HID]
  float* o5  = o4 + SOC_N;          // all_user_social[users]
  float* o6  = o5 + GB;             // all_item_social[pos] = item1_embs[pos]
  float* o7  = o6 + GB;             // all_item_social[neg]
  float* o8  = o7 + GB;             // all_user_rating[users]
  float* o9  = o8 + GB;             // all_item_rating[pos]
  float* o10 = o9 + GB;             // all_item_rating[neg]

  const int gblk = GB / 256;  // 2048

  // all_user = 0.5*(accS/4) + 0.5*(accA/4) = 0.125*(accS+accA)
  sgr_gather2<<<gblk, 256, 0, stream>>>(o0, socAccA, uiAccA, users, 0.5f * invL);
  sgr_gather <<<gblk, 256, 0, stream>>>(o1, uiAccA, pos, N_USERS, invL);
  sgr_gather <<<gblk, 256, 0, stream>>>(o2, uiAccA, neg, N_USERS, invL);
  sgr_scale_copy<<<(SOC_N + 255) / 256, 256, 0, stream>>>(o3, socAccA, invL, SOC_N);
  sgr_scale_copy<<<(SOC_N + 255) / 256, 256, 0, stream>>>(o4, uiAccA, invL, SOC_N);
  sgr_gather <<<gblk, 256, 0, stream>>>(o5, socAccB, users, 0, invL);
  sgr_gather <<<gblk, 256, 0, stream>>>(o6, iemb1, pos, 0, 1.0f);
  sgr_gather <<<gblk, 256, 0, stream>>>(o7, iemb1, neg, 0, 1.0f);
  sgr_gather <<<gblk, 256, 0, stream>>>(o8, uiAccR, users, 0, invL);
  sgr_gather <<<gblk, 256, 0, stream>>>(o9, uiAccR, pos, N_USERS, invL);
  sgr_gather <<<gblk, 256, 0, stream>>>(o10, uiAccR, neg, N_USERS, invL);
}